// SESNetwork_21165598835437
// MI455X (gfx1250) — compile-verified
//
#include <hip/hip_runtime.h>

#define SEN    4096
#define LECN   2048
#define MECN   1024
#define PFCN   4096
#define HPCN   3072
#define TSTEPS 32
#define SLEEPN 32
#define PCIT   5
#define K_LEC  102
#define K_MEC  51
#define K_PFC  204
#define K_HPC  153
#define LMB    0.01f
#define CAPV   20.0f
#define BETA   0.8187307530779818f   /* exp(-1/5) */
#define CDIV(a,b) (((a)+(b)-1)/(b))

typedef __attribute__((ext_vector_type(2))) float v2f;
typedef __attribute__((ext_vector_type(8))) float v8f;

// ---------------- WMMA f32 GEMM (one 16x16 tile per wave32) ----------------
// C[M,N] = scale * A[M,K] @ B[K,N]  (row-major, optional diag zeroing)
__global__ __launch_bounds__(32)
void gemm_nn(const float* __restrict__ A, const float* __restrict__ B,
             float* __restrict__ C, int K, int lda, int ldb, int ldc,
             float scale, int zero_diag) {
  const int tn = blockIdx.x * 16, tm = blockIdx.y * 16;
  const int lane = threadIdx.x & 31, half = lane >> 4, l15 = lane & 15;
  const float* Ar = A + (size_t)(tm + l15) * lda;
  v8f acc = {};
  for (int k = 0; k < K; k += 4) {
    int ka = k + half * 2;
    v2f a, b;
    a.x = Ar[ka];
    a.y = Ar[ka + 1];
    b.x = B[(size_t)ka * ldb + tn + l15];
    b.y = B[(size_t)(ka + 1) * ldb + tn + l15];
    acc = __builtin_amdgcn_wmma_f32_16x16x4_f32(false, a, false, b, (short)0,
                                                acc, false, false);
  }
#pragma unroll
  for (int j = 0; j < 8; ++j) {
    int r = tm + j + half * 8, c = tn + l15;
    float v = acc[j] * scale;
    if (zero_diag && r == c) v = 0.f;
    C[(size_t)r * ldc + c] = v;
  }
}

// C[M,N] = scale * A[M,K] @ Bt[N,K]^T  (Bt row-major N x K)
__global__ __launch_bounds__(32)
void gemm_nt(const float* __restrict__ A, const float* __restrict__ Bt,
             float* __restrict__ C, int K, int lda, int ldb, int ldc,
             float scale, int zero_diag) {
  const int tn = blockIdx.x * 16, tm = blockIdx.y * 16;
  const int lane = threadIdx.x & 31, half = lane >> 4, l15 = lane & 15;
  const float* Ar = A + (size_t)(tm + l15) * lda;
  const float* Br = Bt + (size_t)(tn + l15) * ldb;
  v8f acc = {};
  for (int k = 0; k < K; k += 4) {
    int ka = k + half * 2;
    v2f a, b;
    a.x = Ar[ka];  a.y = Ar[ka + 1];
    b.x = Br[ka];  b.y = Br[ka + 1];
    acc = __builtin_amdgcn_wmma_f32_16x16x4_f32(false, a, false, b, (short)0,
                                                acc, false, false);
  }
#pragma unroll
  for (int j = 0; j < 8; ++j) {
    int r = tm + j + half * 8, c = tn + l15;
    float v = acc[j] * scale;
    if (zero_diag && r == c) v = 0.f;
    C[(size_t)r * ldc + c] = v;
  }
}

// ---------------- exact kwta: radix-select + index-ordered ties ------------
__global__ __launch_bounds__(256)
void kwta_kernel(const float* __restrict__ in, float* __restrict__ out,
                 int n, int k, int istride, int ostride) {
  __shared__ unsigned keys[4096];
  __shared__ unsigned hist[256];
  __shared__ unsigned sh_sel, sh_rem;
  const int col = blockIdx.x;
  const int tid = threadIdx.x, nt = blockDim.x;
  for (int i = tid; i < n; i += nt) {
    unsigned u = __float_as_uint(in[(size_t)i * istride + col]);
    keys[i] = (u & 0x80000000u) ? ~u : (u | 0x80000000u);  // order-preserving
  }
  if (tid == 0) sh_rem = (unsigned)k;
  __syncthreads();
  unsigned pfx = 0;
  for (int p = 0; p < 4; ++p) {
    int shift = 24 - 8 * p;
    for (int b = tid; b < 256; b += nt) hist[b] = 0u;
    __syncthreads();
    for (int i = tid; i < n; i += nt) {
      unsigned key = keys[i];
      bool match = (p == 0) || ((key >> (shift + 8)) == (pfx >> (shift + 8)));
      if (match) atomicAdd(&hist[(key >> shift) & 255u], 1u);
    }
    __syncthreads();
    if (tid == 0) {
      unsigned cum = 0, rem = sh_rem, sel = 0;
      for (int b = 255; b >= 0; --b) {
        unsigned c = hist[b];
        if (cum + c >= rem) { sel = (unsigned)b; sh_rem = rem - cum; break; }
        cum += c;
      }
      sh_sel = sel;
    }
    __syncthreads();
    pfx |= sh_sel << shift;
  }
  const unsigned T = pfx, need_eq = sh_rem;
  for (int i = tid; i < n; i += nt)
    out[(size_t)i * ostride + col] = (keys[i] > T) ? 1.0f : 0.0f;
  __syncthreads();
  if (tid < 32) {  // wave32 index-ordered tie selection
    unsigned cnt = 0;
    for (int base = 0; base < n; base += 32) {
      int i = base + tid;
      bool pred = (keys[i] == T);
      unsigned mask = (unsigned)__ballot(pred);
      unsigned before = cnt + (unsigned)__popc(mask & ((1u << tid) - 1u));
      if (pred && before < need_eq) out[(size_t)i * ostride + col] = 1.0f;
      cnt += (unsigned)__popc(mask);
      if (cnt >= need_eq) break;
    }
  }
}

// ---------------- matvec: out = A @ v (+ addv), wave-per-row ---------------
__global__ __launch_bounds__(256)
void matvec_kernel(const float* __restrict__ A, int lda,
                   const float* __restrict__ v, int vstride,
                   const float* __restrict__ addv,
                   float* __restrict__ out, int cols) {
  int wave = threadIdx.x >> 5, lane = threadIdx.x & 31;
  int row = blockIdx.x * 8 + wave;
  const float* Ar = A + (size_t)row * lda;
  float s = 0.f;
  for (int c = lane; c < cols; c += 32) s += Ar[c] * v[(size_t)c * vstride];
  for (int off = 16; off > 0; off >>= 1) s += __shfl_down(s, off, 32);
  if (lane == 0) out[row] = s + (addv ? addv[row] : 0.f);
}

// ---------------- hebbian add (+diag zero) + row sums ----------------------
__global__ __launch_bounds__(256)
void hebb_rowsum(float* __restrict__ W, int N, int ldw,
                 const float* __restrict__ u, int us,
                 const float* __restrict__ v, int vs,
                 float lmb, int zero_diag, float* __restrict__ post) {
  __shared__ float red[256];
  int r = blockIdx.x, tid = threadIdx.x;
  float ur = lmb * u[(size_t)r * us];
  float* Wr = W + (size_t)r * ldw;
  float s = 0.f;
  for (int c = tid; c < N; c += blockDim.x) {
    float w = Wr[c] + ur * v[(size_t)c * vs];
    if (zero_diag && c == r) w = 0.f;
    Wr[c] = w;
    s += w;
  }
  red[tid] = s; __syncthreads();
  for (int st = 128; st > 0; st >>= 1) {
    if (tid < st) red[tid] += red[tid + st];
    __syncthreads();
  }
  if (tid == 0) post[r] = red[0];
}

// column sums of row-rescaled W (renorm second pass input)
__global__ __launch_bounds__(256)
void colsum_scaled(const float* __restrict__ W, int M, int N, int ldw,
                   const float* __restrict__ post, float* __restrict__ colsum) {
  int c = blockIdx.x * blockDim.x + threadIdx.x;
  if (c >= N) return;
  int rpp = M / gridDim.y;
  int r0 = blockIdx.y * rpp;
  float s = 0.f;
  for (int r = r0; r < r0 + rpp; ++r) {
    float p = post[r];
    float sc = (p > CAPV) ? (CAPV / p) : 1.f;
    s += W[(size_t)r * ldw + c] * sc;
  }
  atomicAdd(&colsum[c], s);
}

__global__ __launch_bounds__(256)
void scale_rc(float* __restrict__ W, int N, int ldw,
              const float* __restrict__ post, const float* __restrict__ colsum) {
  int r = blockIdx.x, tid = threadIdx.x;
  float p = post[r];
  float rs = (p > CAPV) ? (CAPV / p) : 1.f;
  float* Wr = W + (size_t)r * ldw;
  for (int c = tid; c < N; c += blockDim.x) {
    float q = colsum[c];
    float cs = (q > CAPV) ? (CAPV / q) : 1.f;
    Wr[c] *= rs * cs;
  }
}

// sparse-row rank-1: W += lmb * u vT (skip rows where u==0)
__global__ __launch_bounds__(256)
void rank1_rowgate(float* __restrict__ W, int N, int ldw,
                   const float* __restrict__ u, int us,
                   const float* __restrict__ v, int vs, float lmb) {
  int r = blockIdx.x;
  float ur = u[(size_t)r * us];
  if (ur == 0.f) return;
  float s = lmb * ur;
  float* Wr = W + (size_t)r * ldw;
  for (int c = threadIdx.x; c < N; c += blockDim.x)
    Wr[c] += s * v[(size_t)c * vs];
}

// ---------------- small elementwise helpers --------------------------------
__global__ void zero_kernel(float* p, int n) {
  int i = blockIdx.x * blockDim.x + threadIdx.x; if (i < n) p[i] = 0.f;
}
__global__ void copy_kernel(float* o, const float* i, int n) {
  int x = blockIdx.x * blockDim.x + threadIdx.x; if (x < n) o[x] = i[x];
}
__global__ void transpose_cb(float* o, const float* in, int n, int B) {
  int idx = blockIdx.x * blockDim.x + threadIdx.x;
  if (idx < n * B) { int r = idx / B, c = idx % B; o[(size_t)r * B + c] = in[(size_t)c * n + r]; }
}
__global__ void blend_kernel(float* h, const float* b, int n, float beta, float coef) {
  int i = blockIdx.x * blockDim.x + threadIdx.x;
  if (i < n) h[i] = beta * h[i] + coef * b[i];
}
__global__ void addscaled_kernel(float* o, const float* a, const float* noise,
                                 const float* stdp, int n) {
  int i = blockIdx.x * blockDim.x + threadIdx.x;
  if (i < n) o[i] = a[i] + noise[i] * stdp[0];
}
__global__ __launch_bounds__(256)
void minabs_kernel(const float* __restrict__ x, float* stdp, int n) {
  __shared__ float red[256];
  int tid = threadIdx.x;
  float m = 3.402823466e38f;
  for (int i = tid; i < n; i += 256) {
    float a = fabsf(x[i]);
    if (x[i] != 0.f && a < m) m = a;
  }
  red[tid] = m; __syncthreads();
  for (int st = 128; st > 0; st >>= 1) {
    if (tid < st && red[tid + st] < red[tid]) red[tid] = red[tid + st];
    __syncthreads();
  }
  if (tid == 0) stdp[0] = red[0] * 0.1f;
}
__global__ void build_hday(float* Hday, const float* MecBin) {
  int idx = blockIdx.x * blockDim.x + threadIdx.x;
  if (idx < MECN * 32) {
    int i = idx / 32, t = idx % 32;
    Hday[(size_t)(LECN + i) * 32 + t] = (t == 0) ? 0.f : MecBin[(size_t)i * 32 + (t - 1)];
  }
}

// ---------------------------------------------------------------------------
extern "C" void kernel_launch(void* const* d_in, const int* in_sizes, int n_in,
                              void* d_out, int out_size, void* d_ws, size_t ws_size,
                              hipStream_t stream) {
  const float* input    = (const float*)d_in[0];  // (32,4096)
  const float* lec_sen  = (const float*)d_in[1];  // (2048,4096)
  const float* pfc_init = (const float*)d_in[2];  // (4096,4096)
  const float* mec_init = (const float*)d_in[3];  // (1024,4096)
  const float* noise    = (const float*)d_in[4];  // (32,4096)
  const float* h0_hpc   = (const float*)d_in[5];  // (32,3072)
  const float* h0_pfc   = (const float*)d_in[6];  // (32,4096)
  float* pfc_pfc = (float*)d_out;                 // evolving pfc_pfc == output
  (void)in_sizes; (void)n_in; (void)out_size; (void)ws_size;

  float* ws = (float*)d_ws;
  size_t off = 0;
  auto alloc = [&](size_t n) { float* p = ws + off; off += n; return p; };
  float* W_pl   = alloc((size_t)PFCN * LECN);   // pfc_lec
  float* hpcW   = alloc((size_t)HPCN * HPCN);   // hpc_hpc
  float* mecW   = alloc((size_t)MECN * PFCN);   // mec_pfc (working)
  float* LecPre = alloc((size_t)LECN * 32);
  float* LecBin = alloc((size_t)LECN * 32);
  float* Ppre   = alloc((size_t)PFCN * 32);
  float* Hb     = alloc((size_t)PFCN * 32);
  float* Zb     = alloc((size_t)PFCN * 32);
  float* BinB   = alloc((size_t)PFCN * 32);
  float* PfcDay = alloc((size_t)PFCN * 32);
  float* MecPre = alloc((size_t)MECN * 32);
  float* MecBin = alloc((size_t)MECN * 32);
  float* Hday   = alloc((size_t)HPCN * 32);
  float* Hs     = alloc((size_t)HPCN * 32);
  float* Zs     = alloc((size_t)HPCN * 32);
  float* BinS   = alloc((size_t)HPCN * 32);
  float* HpcS   = alloc((size_t)HPCN * 32);
  float* PFCsP  = alloc((size_t)PFCN * 32);
  float* PfcS   = alloc((size_t)PFCN * 32);
  float* hat  = alloc(PFCN); float* tmpv = alloc(PFCN);
  float* hv   = alloc(PFCN); float* zv   = alloc(PFCN);
  float* binv = alloc(PFCN); float* pfcv = alloc(PFCN);
  float* z2   = alloc(MECN); float* mecv = alloc(MECN);
  float* post = alloc(PFCN); float* csum = alloc(PFCN);
  float* stdp = alloc(32);

  // ---- init working state (ws/d_out persist; must re-init every call) ----
  copy_kernel<<<CDIV(PFCN * PFCN, 256), 256, 0, stream>>>(pfc_pfc, pfc_init, PFCN * PFCN);
  copy_kernel<<<CDIV(MECN * PFCN, 256), 256, 0, stream>>>(mecW, mec_init, MECN * PFCN);
  zero_kernel<<<CDIV(PFCN * LECN, 256), 256, 0, stream>>>(W_pl, PFCN * LECN);

  // ---- DAY: batched lec activations (WMMA GEMM 2048x32x4096) ----
  gemm_nt<<<dim3(2, LECN / 16), 32, 0, stream>>>(lec_sen, input, LecPre,
                                                 SEN, SEN, SEN, 32, 1.f, 0);
  kwta_kernel<<<32, 256, 0, stream>>>(LecPre, LecBin, LECN, K_LEC, 32, 32);

  // sequential pfc_lec chain (cheap per-step: matvec + kwta + rank-1)
  for (int t = 0; t < TSTEPS; ++t) {
    matvec_kernel<<<PFCN / 8, 256, 0, stream>>>(W_pl, LECN, LecBin + t, 32,
                                                input + (size_t)t * SEN, hat, LECN);
    minabs_kernel<<<1, 256, 0, stream>>>(hat, stdp, PFCN);
    addscaled_kernel<<<CDIV(PFCN, 256), 256, 0, stream>>>(tmpv, hat,
                                                          noise + (size_t)t * PFCN, stdp, PFCN);
    kwta_kernel<<<1, 256, 0, stream>>>(tmpv, Ppre + t, PFCN, K_PFC, 1, 32);
    rank1_rowgate<<<PFCN, 256, 0, stream>>>(W_pl, LECN, LECN, Ppre + t, 32,
                                            LecBin + t, 32, LMB);
  }

  // batched day pattern completion with fixed pfc_pfc_init (WMMA 4096x32x4096 x5)
  copy_kernel<<<CDIV(PFCN * 32, 256), 256, 0, stream>>>(Hb, Ppre, PFCN * 32);
  for (int it = 0; it < PCIT; ++it) {
    gemm_nn<<<dim3(2, PFCN / 16), 32, 0, stream>>>(pfc_init, Hb, Zb,
                                                   PFCN, PFCN, 32, 32, 1.f, 0);
    kwta_kernel<<<32, 256, 0, stream>>>(Zb, BinB, PFCN, K_PFC, 32, 32);
    blend_kernel<<<CDIV(PFCN * 32, 256), 256, 0, stream>>>(Hb, BinB, PFCN * 32,
                                                           BETA, 1.f - BETA);
  }
  kwta_kernel<<<32, 256, 0, stream>>>(Hb, PfcDay, PFCN, K_PFC, 32, 32);
  gemm_nn<<<dim3(2, MECN / 16), 32, 0, stream>>>(mec_init, PfcDay, MecPre,
                                                 PFCN, PFCN, 32, 32, 1.f, 0);
  kwta_kernel<<<32, 256, 0, stream>>>(MecPre, MecBin, MECN, K_MEC, 32, 32);

  // hpc_hpc = LMB * Hday Hday^T, zero diag (WMMA 3072x3072x32)
  copy_kernel<<<CDIV(LECN * 32, 256), 256, 0, stream>>>(Hday, LecBin, LECN * 32);
  build_hday<<<CDIV(MECN * 32, 256), 256, 0, stream>>>(Hday, MecBin);
  gemm_nt<<<dim3(HPCN / 16, HPCN / 16), 32, 0, stream>>>(Hday, Hday, hpcW,
                                                         32, 32, 32, HPCN, LMB, 1);

  // ---- SLEEP: batched hpc pattern completion + pfc projection ----
  transpose_cb<<<CDIV(HPCN * 32, 256), 256, 0, stream>>>(Hs, h0_hpc, HPCN, 32);
  for (int it = 0; it < PCIT; ++it) {
    gemm_nn<<<dim3(2, HPCN / 16), 32, 0, stream>>>(hpcW, Hs, Zs,
                                                   HPCN, HPCN, 32, 32, 1.f, 0);
    kwta_kernel<<<32, 256, 0, stream>>>(Zs, BinS, HPCN, K_HPC, 32, 32);
    blend_kernel<<<CDIV(HPCN * 32, 256), 256, 0, stream>>>(Hs, BinS, HPCN * 32,
                                                           BETA, 1.f);  // no (1-beta)
  }
  kwta_kernel<<<32, 256, 0, stream>>>(Hs, HpcS, HPCN, K_HPC, 32, 32);
  gemm_nn<<<dim3(2, PFCN / 16), 32, 0, stream>>>(W_pl, HpcS, PFCsP,
                                                 LECN, LECN, 32, 32, 1.f, 0);
  kwta_kernel<<<32, 256, 0, stream>>>(PFCsP, PfcS, PFCN, K_PFC, 32, 32);

  // sequential sleep consolidation
  for (int s = 0; s < SLEEPN; ++s) {
    // pfc_pfc: hebbian + diag zero + renorm (L2-resident passes)
    zero_kernel<<<CDIV(PFCN, 256), 256, 0, stream>>>(csum, PFCN);
    hebb_rowsum<<<PFCN, 256, 0, stream>>>(pfc_pfc, PFCN, PFCN, PfcS + s, 32,
                                          PfcS + s, 32, LMB, 1, post);
    colsum_scaled<<<dim3(PFCN / 256, 8), 256, 0, stream>>>(pfc_pfc, PFCN, PFCN, PFCN,
                                                           post, csum);
    scale_rc<<<PFCN, 256, 0, stream>>>(pfc_pfc, PFCN, PFCN, post, csum);
    // pc_pfc(h0p, pfc_pfc, 50)
    copy_kernel<<<CDIV(PFCN, 256), 256, 0, stream>>>(hv, h0_pfc + (size_t)s * PFCN, PFCN);
    for (int it = 0; it < PCIT; ++it) {
      matvec_kernel<<<PFCN / 8, 256, 0, stream>>>(pfc_pfc, PFCN, hv, 1, nullptr, zv, PFCN);
      kwta_kernel<<<1, 256, 0, stream>>>(zv, binv, PFCN, 50, 1, 1);
      blend_kernel<<<CDIV(PFCN, 256), 256, 0, stream>>>(hv, binv, PFCN, BETA, 1.f - BETA);
    }
    kwta_kernel<<<1, 256, 0, stream>>>(hv, pfcv, PFCN, 50, 1, 1);
    // mec_s = kwta(mec_pfc @ pfc, 5)
    matvec_kernel<<<MECN / 8, 256, 0, stream>>>(mecW, PFCN, pfcv, 1, nullptr, z2, PFCN);
    kwta_kernel<<<1, 256, 0, stream>>>(z2, mecv, MECN, 5, 1, 1);
    // mec_pfc: hebbian + renorm (no diag mask)
    zero_kernel<<<CDIV(PFCN, 256), 256, 0, stream>>>(csum, PFCN);
    hebb_rowsum<<<MECN, 256, 0, stream>>>(mecW, PFCN, PFCN, mecv, 1, pfcv, 1,
                                          LMB, 0, post);
    colsum_scaled<<<dim3(PFCN / 256, 8), 256, 0, stream>>>(mecW, MECN, PFCN, PFCN,
                                                           post, csum);
    scale_rc<<<MECN, 256, 0, stream>>>(mecW, PFCN, PFCN, post, csum);
  }
  // d_out already holds the final pfc_pfc
}